// HarmonicOscillatorWithInteraction1D_42923903156649
// MI455X (gfx1250) — compile-verified
//
#include <hip/hip_runtime.h>
#include <math.h>

// MI455X / gfx1250, wave32. One wave per walker (B=4096, A=16, H=256).
//
// Math: kin = -0.5 * tr(M^{-1} M2)   (grad^2 terms cancel analytically)
//   M [a,j] = sum_k tanh(x_a w_k + c_k)          * W2[k,j] + b2[j]
//   M2[a,j] = sum_k (-2 h (1-h^2) w_k^2)         * W2[k,j]
// energy = kin + 0.5*sum(x^2) + GCONST * sum_{i<j} exp(-2 (x_j-x_i)^2)

typedef __attribute__((ext_vector_type(2))) float v2f;
typedef __attribute__((ext_vector_type(8))) float v8f;
typedef unsigned int v4ui __attribute__((ext_vector_type(4)));
typedef int          v8si __attribute__((ext_vector_type(8)));
typedef int          v4si __attribute__((ext_vector_type(4)));

#define A_DIM 16
#define H_DIM 256
#define WPB   8     // waves per block
#define BLK   256

// -20 / (sqrt(2*pi) * 0.5)
#define GCONST (-15.957691216057308f)

#if defined(__gfx1250__) && __has_builtin(__builtin_amdgcn_tensor_load_to_lds) && \
    __has_builtin(__builtin_amdgcn_s_wait_tensorcnt)
#define USE_TDM 1
#else
#define USE_TDM 0
#endif

// CDNA5 hardware tanh (TRANS op). Inline-asm fallback carries its own v_nop
// to satisfy the TRANS32 RAW hazard (hazard recognizer can't see into asm).
__device__ __forceinline__ float fast_tanh(float u) {
#if __has_builtin(__builtin_amdgcn_tanhf)
  return __builtin_amdgcn_tanhf(u);
#elif __has_builtin(__builtin_amdgcn_tanh_f32)
  return __builtin_amdgcn_tanh_f32(u);
#else
  float r;
  asm("v_tanh_f32 %0, %1\n\tv_nop" : "=v"(r) : "v"(u));
  return r;
#endif
}

__global__ __launch_bounds__(BLK) void vmc_local_energy_kernel(
    const float* __restrict__ x, const float* __restrict__ W1,
    const float* __restrict__ b1, const float* __restrict__ W2,
    const float* __restrict__ b2, float* __restrict__ out)
{
  __shared__ float sW1[H_DIM];
  __shared__ float sB1[H_DIM];
  __shared__ float sW2[H_DIM * A_DIM];
  __shared__ float sB2[A_DIM];
  __shared__ float sM[WPB][A_DIM][A_DIM];
  __shared__ float sG[WPB][A_DIM][A_DIM];

  const int tid = threadIdx.x;

  // ---- stage weights in LDS (shared by the 8 walkers of this block) ----
  sW1[tid] = W1[tid];
  sB1[tid] = b1[tid];
  if (tid < A_DIM) sB2[tid] = b2[tid];

#if USE_TDM
  // W2 (256x16 f32, 16 KB) staged by the Tensor Data Mover.
  if (tid == 0) {
    const unsigned long long ga = (unsigned long long)(const void*)W2;
    const unsigned lds_off = (unsigned)(unsigned long long)(void*)&sW2[0];
    // D# group 0: count=1 | lds_addr | global_addr[56:0] | type=2
    v4ui g0 = { 1u,
                lds_off,
                (unsigned)(ga & 0xFFFFFFFFu),
                (unsigned)((ga >> 32) & 0x01FFFFFFu) | (2u << 30) };
    // D# group 1: data_size=4B; tensor_dim0=16, tensor_dim1=256;
    // tile_dim0=16, tile_dim1=256; tensor_dim0_stride=16.
    v8si g1 = { (int)(2u << 16),          // [17:16] data_size = 2 (4 bytes)
                (int)(16u << 16),         // tensor_dim0[15:0] @ bits 63:48
                (int)(256u << 16),        // dim0 hi16 = 0 | tensor_dim1[15:0]
                (int)(16u << 16),         // dim1 hi16 = 0 | tile_dim0 = 16
                (int)256,                 // tile_dim1 = 256 | tile_dim2 = 0
                (int)16,                  // tensor_dim0_stride[31:0] = 16
                0, 0 };
    v4si gz = { 0, 0, 0, 0 };
#if __clang_major__ >= 23
    v8si gz8 = { 0, 0, 0, 0, 0, 0, 0, 0 };
    __builtin_amdgcn_tensor_load_to_lds(g0, g1, gz, gz, gz8, 0);
#else
    __builtin_amdgcn_tensor_load_to_lds(g0, g1, gz, gz, 0);
#endif
  }
  __builtin_amdgcn_s_wait_tensorcnt(0);
#else
#pragma unroll
  for (int i = 0; i < (H_DIM * A_DIM) / BLK; ++i)
    sW2[i * BLK + tid] = W2[i * BLK + tid];
#endif
  __syncthreads();

  const int wave   = tid >> 5;
  const int lane   = tid & 31;
  const int walker = blockIdx.x * WPB + wave;
  const int a  = lane & 15;   // A-row / B-col / C-col index
  const int hi = lane >> 4;   // half-wave selects K pair

  const float xa = x[walker * A_DIM + a];

  // ---- two 16x256x16 f32 GEMMs via V_WMMA_F32_16X16X4_F32 ----
  v8f accM = {0.f,0.f,0.f,0.f,0.f,0.f,0.f,0.f};
  v8f accG = {0.f,0.f,0.f,0.f,0.f,0.f,0.f,0.f};

#pragma unroll 4
  for (int kk = 0; kk < H_DIM / 4; ++kk) {
    const int k = kk * 4 + hi * 2;
    const float w0 = sW1[k], w1 = sW1[k + 1];
    const float c0 = sB1[k], c1 = sB1[k + 1];
    const float h0 = fast_tanh(fmaf(xa, w0, c0));
    const float h1 = fast_tanh(fmaf(xa, w1, c1));
    // A tiles: 16x4 f32 layout -> lane = row, {K=2*hi, 2*hi+1} per lane
    v2f aH = { h0, h1 };
    v2f aG = { -2.f * h0 * (1.f - h0 * h0) * w0 * w0,
               -2.f * h1 * (1.f - h1 * h1) * w1 * w1 };
    // B tile: 4x16 f32, lane = col, rows {k, k+1}
    v2f bW = { sW2[k * A_DIM + a], sW2[(k + 1) * A_DIM + a] };

    accM = __builtin_amdgcn_wmma_f32_16x16x4_f32(
        false, aH, false, bW, (short)0, accM, false, false);
    accG = __builtin_amdgcn_wmma_f32_16x16x4_f32(
        false, aG, false, bW, (short)0, accG, false, false);
  }

  // add b2[col] to M (C/D layout: VGPR r -> row r + 8*hi, col = lane&15)
  const float b2v = sB2[a];
#pragma unroll
  for (int r = 0; r < 8; ++r) accM[r] += b2v;

  // ---- transpose C/D tiles through LDS into column-per-lane layout ----
#pragma unroll
  for (int r = 0; r < 8; ++r) {
    sM[wave][r + 8 * hi][a] = accM[r];
    sG[wave][r + 8 * hi][a] = accG[r];
  }
  __syncthreads();

  // lanes 0..15 own columns of M; lanes 16..31 own columns of M2
  float col[16];
#pragma unroll
  for (int r = 0; r < 16; ++r)
    col[r] = (hi == 0) ? sM[wave][r][a] : sG[wave][r][a];

  // ---- Gauss-Jordan with partial pivoting: [M | M2] -> [I | M^{-1} M2] ----
#pragma unroll
  for (int p = 0; p < 16; ++p) {
    // pivot row chosen by lane p (owner of column p of M)
    float bestv = fabsf(col[p]);
    int   besti = p;
#pragma unroll
    for (int r = p + 1; r < 16; ++r) {
      float v = fabsf(col[r]);
      if (v > bestv) { bestv = v; besti = r; }
    }
    const int piv = __shfl(besti, p, 32);

    // swap rows p <-> piv in every lane's column
    float vp = col[p], vpiv = vp;
#pragma unroll
    for (int r = 0; r < 16; ++r) if (r == piv) vpiv = col[r];
#pragma unroll
    for (int r = 0; r < 16; ++r) if (r == piv) col[r] = vp;
    col[p] = vpiv;

    // scale row p by 1/pivot
    const float pvv = __shfl(col[p], p, 32);
    col[p] *= (1.0f / pvv);

    // broadcast (scaled) column p, then eliminate
    float brd[16];
#pragma unroll
    for (int r = 0; r < 16; ++r) brd[r] = __shfl(col[r], p, 32);
#pragma unroll
    for (int r = 0; r < 16; ++r)
      if (r != p) col[r] = fmaf(-brd[r], col[p], col[r]);
  }

  // diagonal of Y = M^{-1} M2 (held by lanes 16..31, col index a = lane-16)
  float ydiag = 0.f;
#pragma unroll
  for (int r = 0; r < 16; ++r) if (r == a) ydiag = col[r];

  // pairwise Gaussian row-sum (meaningful on lanes 0..15)
  float gsum = 0.f;
#pragma unroll
  for (int j = 0; j < 16; ++j) {
    const float xj = __shfl(xa, j, 32);
    const float d  = xj - xa;
    if (j > a) gsum += expf(-2.0f * d * d);   // sigma0=0.5 -> 1/(2s^2)=2
  }

  // lanes 0..15: potential + interaction; lanes 16..31: -0.5 * tr term
  float partial = (hi == 0) ? fmaf(0.5f * xa, xa, GCONST * gsum)
                            : (-0.5f * ydiag);
#pragma unroll
  for (int off = 16; off > 0; off >>= 1)
    partial += __shfl_xor(partial, off, 32);

  if (lane == 0) out[walker] = partial;
}

extern "C" void kernel_launch(void* const* d_in, const int* in_sizes, int n_in,
                              void* d_out, int out_size, void* d_ws, size_t ws_size,
                              hipStream_t stream) {
  const float* x  = (const float*)d_in[0];  // [B, 16]
  const float* W1 = (const float*)d_in[1];  // [1, 256]
  const float* b1 = (const float*)d_in[2];  // [256]
  const float* W2 = (const float*)d_in[3];  // [256, 16]
  const float* b2 = (const float*)d_in[4];  // [16]
  float* out = (float*)d_out;               // [B]

  const int B = in_sizes[0] / A_DIM;        // 4096 walkers
  const int blocks = B / WPB;               // one wave32 per walker
  vmc_local_energy_kernel<<<blocks, BLK, 0, stream>>>(x, W1, b1, W2, b2, out);
}